// GraphMultiAttentionV2_24558622998901
// MI455X (gfx1250) — compile-verified
//
#include <hip/hip_runtime.h>

#define NN 50000
#define EE 800000
#define FD 128
#define HH 8
#define DH 16

typedef __attribute__((ext_vector_type(2))) float v2f;
typedef __attribute__((ext_vector_type(4))) float v4f;
typedef __attribute__((ext_vector_type(8))) float v8f;

__device__ __forceinline__ void atomic_max_f32(float* addr, float val) {
    if (val >= 0.0f) atomicMax((int*)addr, __float_as_int(val));
    else             atomicMin((unsigned int*)addr, __float_as_uint(val));
}

// ---------------------------------------------------------------------------
// K0: initialize agg (N*F zeros), m (N*H -inf), s (N*H zeros)
// ---------------------------------------------------------------------------
__global__ void init_kernel(float* __restrict__ agg, float* __restrict__ m,
                            float* __restrict__ s) {
    size_t idx = (size_t)blockIdx.x * blockDim.x + threadIdx.x;
    if (idx < (size_t)NN * FD) agg[idx] = 0.0f;
    if (idx < (size_t)NN * HH) {
        m[idx] = __int_as_float(0xff800000);  // -inf
        s[idx] = 0.0f;
    }
}

// ---------------------------------------------------------------------------
// Shared WMMA tile: out[16 rows, 16 cols] = A[16,128] @ W^T[128,16-col-tile]
// A lane layout (f32 16x16x4): lane l -> row l&15, K-pair offset (l>>4)*2
// B = W^T so B[k][c] = W[c][k]: lane reads W row (c0 + (l&15)), contiguous K.
// C layout: vgpr r, lane l -> M = r + (l>=16 ? 8 : 0), Ncol = l&15.
// ---------------------------------------------------------------------------
__device__ __forceinline__ v8f wmma_tile_f32(const float* __restrict__ arow,
                                             const float* __restrict__ brow) {
    v8f acc = {};
#pragma unroll
    for (int k0 = 0; k0 < FD; k0 += 4) {
        v2f a = *(const v2f*)(arow + k0);
        v2f b = *(const v2f*)(brow + k0);
        acc = __builtin_amdgcn_wmma_f32_16x16x4_f32(false, a, false, b,
                                                    (short)0, acc, false, false);
    }
    return acc;
}

// ---------------------------------------------------------------------------
// K1: q/k/v projections.  One wave = (node-tile, matrix, col-tile).
//     waves = (NN/16) * 3 * 8 = 75000
// ---------------------------------------------------------------------------
__global__ void qkv_proj_kernel(const float* __restrict__ feat,
                                const float* __restrict__ Wq,
                                const float* __restrict__ Wk,
                                const float* __restrict__ Wv,
                                float* __restrict__ q, float* __restrict__ k,
                                float* __restrict__ v) {
    int wave = (int)((blockIdx.x * blockDim.x + threadIdx.x) >> 5);
    int lane = threadIdx.x & 31;
    const int WAVES = (NN / 16) * 24;
    if (wave >= WAVES) return;
    int tile = wave / 24;
    int t    = wave % 24;
    int mat  = t >> 3;
    int ct   = t & 7;
    const float* W   = (mat == 0) ? Wq : (mat == 1) ? Wk : Wv;
    float*       out = (mat == 0) ? q  : (mat == 1) ? k  : v;
    int n0 = tile * 16, c0 = ct * 16;
    int r    = lane & 15;
    int koff = (lane >> 4) * 2;
    const float* arow = feat + (size_t)(n0 + r) * FD + koff;
    const float* brow = W    + (size_t)(c0 + r) * FD + koff;
    v8f acc = wmma_tile_f32(arow, brow);
    int mbase = n0 + ((lane >> 4) << 3);
    int col   = c0 + (lane & 15);
#pragma unroll
    for (int rr = 0; rr < 8; ++rr)
        out[(size_t)(mbase + rr) * FD + col] = acc[rr];
}

// ---------------------------------------------------------------------------
// K2: per 16-edge tile: WMMA bias/gate GEMM -> LDS, then per (edge,head)
//     q.k dot, clamp, bias, scale, sigmoid gate, atomic-max into m[dst].
//     tiles = EE/16 = 50000, 8 waves per 256-thread block.
// ---------------------------------------------------------------------------
__global__ void edge_logits_kernel(const float* __restrict__ edge_feat,
                                   const float* __restrict__ Wedge,
                                   const float* __restrict__ Wgate,
                                   const float* __restrict__ q,
                                   const float* __restrict__ k,
                                   const int* __restrict__ src,
                                   const int* __restrict__ dst,
                                   float* __restrict__ logits,
                                   float* __restrict__ gates,
                                   float* __restrict__ m) {
    __shared__ float lds[8 * 256];
    int wib  = threadIdx.x >> 5;
    int wave = blockIdx.x * 8 + wib;
    int lane = threadIdx.x & 31;
    const int TILES = EE / 16;
    if (wave >= TILES) return;
    int e0   = wave * 16;
    int r    = lane & 15;
    int koff = (lane >> 4) * 2;
    const float* arow = edge_feat + (size_t)(e0 + r) * FD + koff;
    // stream next tile of edge_feat through GL2 (global_prefetch_b8)
    __builtin_prefetch(arow + (size_t)16 * FD, 0, 1);
    int col = lane & 15;
    const float* brow =
        ((col < 8) ? (Wedge + (size_t)col * FD) : (Wgate + (size_t)(col - 8) * FD)) + koff;
    v8f acc = wmma_tile_f32(arow, brow);

    float* myLds = lds + wib * 256;
    int mbase = (lane >> 4) << 3;
#pragma unroll
    for (int rr = 0; rr < 8; ++rr)
        myLds[(mbase + rr) * 16 + col] = acc[rr];
    asm volatile("s_wait_dscnt 0" ::: "memory");  // lockstep wave: DS in-order

#pragma unroll
    for (int i = 0; i < 4; ++i) {
        int p  = lane + 32 * i;      // 16 edges x 8 heads = 128 pairs
        int el = p >> 3;
        int h  = p & 7;
        int e  = e0 + el;
        int sN = src[e], dN = dst[e];
        const float* qp = q + (size_t)sN * FD + h * DH;
        const float* kp = k + (size_t)dN * FD + h * DH;
        float a = 0.0f;
#pragma unroll
        for (int dd = 0; dd < DH; dd += 4) {
            v4f qv = *(const v4f*)(qp + dd);
            v4f kv = *(const v4f*)(kp + dd);
            a += qv.x * kv.x + qv.y * kv.y + qv.z * kv.z + qv.w * kv.w;
        }
        a = fminf(5.0f, fmaxf(-5.0f, a));
        float bias  = myLds[el * 16 + h];
        float gpre  = myLds[el * 16 + 8 + h];
        float logit = (a + bias) * 4.0f;  // / SCALING, SCALING = D^-0.5 = 0.25
        logits[(size_t)e * HH + h] = logit;
        gates[(size_t)e * HH + h]  = 1.0f / (1.0f + __expf(-gpre));
        atomic_max_f32(&m[(size_t)dN * HH + h], logit);
    }
}

// ---------------------------------------------------------------------------
// K3: ex = exp(logit - m[dst]); s[dst] += ex  (logits buffer becomes ex)
// ---------------------------------------------------------------------------
__global__ void softmax_sum_kernel(const int* __restrict__ dst,
                                   const float* __restrict__ m,
                                   float* __restrict__ logits,
                                   float* __restrict__ s) {
    size_t idx = (size_t)blockIdx.x * blockDim.x + threadIdx.x;
    if (idx >= (size_t)EE * HH) return;
    int e  = (int)(idx >> 3);
    int h  = (int)(idx & 7);
    int dN = dst[e];
    float ex = __expf(logits[idx] - m[(size_t)dN * HH + h]);
    logits[idx] = ex;
    unsafeAtomicAdd(&s[(size_t)dN * HH + h], ex);
}

// ---------------------------------------------------------------------------
// K4: agg[dst,h,:] += v[src,h,:] * (ex / s[dst,h] * gate)
// ---------------------------------------------------------------------------
__global__ void aggregate_kernel(const int* __restrict__ src,
                                 const int* __restrict__ dst,
                                 const float* __restrict__ ex,
                                 const float* __restrict__ gates,
                                 const float* __restrict__ s,
                                 const float* __restrict__ v,
                                 float* __restrict__ agg) {
    size_t idx = (size_t)blockIdx.x * blockDim.x + threadIdx.x;
    if (idx >= (size_t)EE * HH) return;
    int e  = (int)(idx >> 3);
    int h  = (int)(idx & 7);
    int sN = src[e], dN = dst[e];
    float sa = ex[idx] / s[(size_t)dN * HH + h] * gates[idx];
    const float* vp = v   + (size_t)sN * FD + h * DH;
    float*       ap = agg + (size_t)dN * FD + h * DH;
#pragma unroll
    for (int dd = 0; dd < DH; ++dd)
        unsafeAtomicAdd(&ap[dd], vp[dd] * sa);
}

// ---------------------------------------------------------------------------
// K5: rst = agg @ Wnode^T.  waves = (NN/16) * 8 = 25000
// ---------------------------------------------------------------------------
__global__ void out_proj_kernel(const float* __restrict__ agg,
                                const float* __restrict__ Wnode,
                                float* __restrict__ out) {
    int wave = (int)((blockIdx.x * blockDim.x + threadIdx.x) >> 5);
    int lane = threadIdx.x & 31;
    const int WAVES = (NN / 16) * 8;
    if (wave >= WAVES) return;
    int tile = wave >> 3;
    int ct   = wave & 7;
    int n0 = tile * 16, c0 = ct * 16;
    int r    = lane & 15;
    int koff = (lane >> 4) * 2;
    const float* arow = agg   + (size_t)(n0 + r) * FD + koff;
    const float* brow = Wnode + (size_t)(c0 + r) * FD + koff;
    v8f acc = wmma_tile_f32(arow, brow);
    int mbase = n0 + ((lane >> 4) << 3);
    int col   = c0 + (lane & 15);
#pragma unroll
    for (int rr = 0; rr < 8; ++rr)
        out[(size_t)(mbase + rr) * FD + col] = acc[rr];
}

// ---------------------------------------------------------------------------
extern "C" void kernel_launch(void* const* d_in, const int* in_sizes, int n_in,
                              void* d_out, int out_size, void* d_ws, size_t ws_size,
                              hipStream_t stream) {
    const float* feat      = (const float*)d_in[0];
    const float* edge_feat = (const float*)d_in[1];
    const float* Wq        = (const float*)d_in[2];
    const float* Wk        = (const float*)d_in[3];
    const float* Wv        = (const float*)d_in[4];
    const float* Wnode     = (const float*)d_in[5];
    const float* Wedge     = (const float*)d_in[6];
    const float* Wgate     = (const float*)d_in[7];
    const int*   src       = (const int*)d_in[8];
    const int*   dst       = (const int*)d_in[9];
    float*       out       = (float*)d_out;

    // workspace layout (floats)
    float* ws = (float*)d_ws;
    float* q      = ws;                 ws += (size_t)NN * FD;
    float* k      = ws;                 ws += (size_t)NN * FD;
    float* v      = ws;                 ws += (size_t)NN * FD;
    float* logits = ws;                 ws += (size_t)EE * HH;   // becomes ex
    float* gates  = ws;                 ws += (size_t)EE * HH;
    float* m      = ws;                 ws += (size_t)NN * HH;
    float* s      = ws;                 ws += (size_t)NN * HH;
    float* agg    = ws;                 ws += (size_t)NN * FD;

    // K0: init (covers NN*FD elements, the largest region)
    {
        size_t n = (size_t)NN * FD;
        init_kernel<<<(unsigned)((n + 255) / 256), 256, 0, stream>>>(agg, m, s);
    }
    // K1: qkv projections — 75000 waves
    {
        int waves = (NN / 16) * 24;
        qkv_proj_kernel<<<(waves * 32 + 255) / 256, 256, 0, stream>>>(
            feat, Wq, Wk, Wv, q, k, v);
    }
    // K2: edge logits + gates + segment max — 50000 tiles, 8 waves/block
    {
        int tiles = EE / 16;
        edge_logits_kernel<<<(tiles + 7) / 8, 256, 0, stream>>>(
            edge_feat, Wedge, Wgate, q, k, src, dst, logits, gates, m);
    }
    // K3: exp + segment sum
    {
        size_t n = (size_t)EE * HH;
        softmax_sum_kernel<<<(unsigned)((n + 255) / 256), 256, 0, stream>>>(
            dst, m, logits, s);
    }
    // K4: weighted scatter aggregation
    {
        size_t n = (size_t)EE * HH;
        aggregate_kernel<<<(unsigned)((n + 255) / 256), 256, 0, stream>>>(
            src, dst, logits, gates, s, v, agg);
    }
    // K5: output projection — 25000 waves
    {
        int waves = (NN / 16) * 8;
        out_proj_kernel<<<(waves * 32 + 255) / 256, 256, 0, stream>>>(
            agg, Wnode, out);
    }
}